// DynamicDWConv_12884901888105
// MI455X (gfx1250) — compile-verified
//
#include <hip/hip_runtime.h>
#include <hip/hip_bf16.h>

#define B_    8
#define DIM_  192
#define H_    128
#define W_    128
#define K_    7
#define HID_  48
#define KK_   49
#define NOUT_ (DIM_ * KK_)   // 9408
#define EPS_  1e-5f

typedef __attribute__((ext_vector_type(2))) float v2f;
typedef __attribute__((ext_vector_type(8))) float v8f;

static __device__ __forceinline__ v2f mk2(float a, float b) {
  v2f r; r.x = a; r.y = b; return r;
}

// ---------------------------------------------------------------------------
// Kernel 1: global average pool over H*W per (b,c) plane.  Pure-bandwidth.
// ---------------------------------------------------------------------------
__global__ void pool_kernel(const float* __restrict__ x, float* __restrict__ pooled) {
  __shared__ float red[256];
  const int bc = blockIdx.x;                       // 0 .. B*DIM-1
  const float* p = x + (size_t)bc * (H_ * W_);
  float acc = 0.f;
  for (int i = threadIdx.x; i < H_ * W_; i += 256) acc += p[i];
  red[threadIdx.x] = acc;
  __syncthreads();
  for (int s = 128; s > 0; s >>= 1) {
    if (threadIdx.x < s) red[threadIdx.x] += red[threadIdx.x + s];
    __syncthreads();
  }
  if (threadIdx.x == 0) pooled[bc] = red[0] * (1.0f / (H_ * W_));
}

// ---------------------------------------------------------------------------
// Kernel 2: FC1 ([8,192] x [48,192]^T) + BatchNorm(eval) + ReLU.
// Writes a ZERO-PADDED y of shape [16, 48] (rows 8..15 = 0) so the WMMA
// kernel can load its A-fragment unconditionally (no exec-mask branches).
// ---------------------------------------------------------------------------
__global__ void se_fc1_kernel(const float* __restrict__ pooled,
                              const float* __restrict__ w1,
                              const float* __restrict__ gamma,
                              const float* __restrict__ beta,
                              const float* __restrict__ mean,
                              const float* __restrict__ var,
                              float* __restrict__ ypad) {
  const int o = blockIdx.x * blockDim.x + threadIdx.x;
  if (o >= 16 * HID_) return;
  const int b = o / HID_;
  const int h = o % HID_;
  if (b >= B_) { ypad[o] = 0.f; return; }
  const float* pv = pooled + b * DIM_;
  const float* wv = w1 + h * DIM_;
  float acc = 0.f;
#pragma unroll 8
  for (int c = 0; c < DIM_; ++c) acc += pv[c] * wv[c];
  float t = (acc - mean[h]) * rsqrtf(var[h] + EPS_) * gamma[h] + beta[h];
  ypad[o] = t > 0.f ? t : 0.f;
}

// ---------------------------------------------------------------------------
// Kernel 3: FC2 via V_WMMA_F32_16X16X4_F32.
//   wdyn[16(M=batch, rows 8..15 zero), 9408(N)] = ypad[16,48] x w2[9408,48]^T + b2
//   One wave per 16-wide N tile; 12 chained K=4 WMMAs.  A and B fragments are
//   8B-aligned float2's -> single global_load_b64 each, no predication.
// A layout: lane L -> row M=L&15; vgpr j holds K = k + j + 2*(L>>4)
// B layout: lane L -> col N=L&15; vgpr j holds K-row k + j + 2*(L>>4)
// D layout: vgpr r, lanes 0..15 -> (M=r, N=lane)  [lanes 16..31 are M=r+8 >= B, dropped]
// ---------------------------------------------------------------------------
__global__ void se_fc2_wmma_kernel(const float* __restrict__ ypad,
                                   const float* __restrict__ w2,
                                   const float* __restrict__ b2,
                                   float* __restrict__ wdyn) {
  const int lane  = threadIdx.x;       // 0..31
  const int row   = lane & 15;
  const int half  = lane >> 4;         // 0 or 1
  const int nbase = blockIdx.x * 16;
  const int n     = nbase + row;

  const float* ap = ypad + row * HID_ + 2 * half;           // + k
  const float* bp = w2 + (size_t)n * HID_ + 2 * half;       // + k

  v8f acc = {};
#pragma unroll
  for (int k = 0; k < HID_; k += 4) {
    const v2f a = *(const v2f*)(ap + k);   // 8B-aligned: k even, row*48 even
    const v2f b = *(const v2f*)(bp + k);
    acc = __builtin_amdgcn_wmma_f32_16x16x4_f32(
        /*neg_a=*/false, a, /*neg_b=*/false, b,
        /*c_mod=*/(short)0, acc, /*reuse_a=*/false, /*reuse_b=*/false);
  }

  if (half == 0) {
    const float bb = b2[n];
#pragma unroll
    for (int r = 0; r < 8; ++r)
      wdyn[(size_t)r * NOUT_ + n] = acc[r] + bb;
  }
}

// ---------------------------------------------------------------------------
// Kernel 4: per-sample depthwise 7x7 conv + bias.  VALU-bound (49 FMA/pixel).
// Block: 256 threads, one (b,c) plane strip of 16 rows.  Input tile staged in
// LDS with 3-pixel halo; 49 dynamic weights held in VGPRs; each thread owns
// TWO adjacent output columns with a float2 accumulator (packed fp32 FMA) and
// slides an 8-float register window across the 7 horizontal taps.
// ---------------------------------------------------------------------------
#define TILE_H 16
#define TROWS  (TILE_H + 6)   // 22
#define TCOLS  134            // 128 + 6 halo
#define TPITCH 136            // pad to dodge bank conflicts

__global__ void dwconv_kernel(const float* __restrict__ x,
                              const float* __restrict__ wdyn,
                              const float* __restrict__ bias,
                              float* __restrict__ out) {
  __shared__ float tile[TROWS * TPITCH];
  __shared__ float wsh[KK_];

  const int bc  = blockIdx.x;              // b*DIM + c
  const int c   = bc % DIM_;
  const int b   = bc / DIM_;
  const int y0  = blockIdx.y * TILE_H;
  const int tid = threadIdx.x;
  const float* xp = x + (size_t)bc * (H_ * W_);

  if (tid < KK_) wsh[tid] = wdyn[(size_t)b * NOUT_ + c * KK_ + tid];

  // Stage input rows [y0-3, y0+18] x cols [-3, 130] (zero-padded OOB).
  for (int idx = tid; idx < TROWS * TCOLS; idx += 256) {
    const int r  = idx / TCOLS;
    const int cc = idx % TCOLS;
    const int gy = y0 - 3 + r;
    const int gx = cc - 3;
    float v = 0.f;
    if ((unsigned)gy < (unsigned)H_ && (unsigned)gx < (unsigned)W_)
      v = xp[gy * W_ + gx];
    tile[r * TPITCH + cc] = v;
  }
  __syncthreads();

  float w[KK_];
#pragma unroll
  for (int i = 0; i < KK_; ++i) w[i] = wsh[i];

  const float bv = bias[c];
  const int px = tid & 63;                 // column pair -> cols 2*px, 2*px+1
  const int rg = tid >> 6;                 // 0..3 -> 4 output rows each
  float* op = out + (size_t)bc * (H_ * W_);

#pragma unroll
  for (int rr = 0; rr < 4; ++rr) {
    const int ry = rg * 4 + rr;            // output row within tile
    v2f acc = mk2(bv, bv);
#pragma unroll
    for (int i = 0; i < K_; ++i) {
      const float* trow = &tile[(ry + i) * TPITCH + 2 * px];
      float e[8];
#pragma unroll
      for (int t = 0; t < 8; ++t) e[t] = trow[t];
#pragma unroll
      for (int j = 0; j < K_; ++j)
        acc += mk2(e[j], e[j + 1]) * w[i * K_ + j];
    }
    op[(y0 + ry) * W_ + 2 * px]     = acc.x;
    op[(y0 + ry) * W_ + 2 * px + 1] = acc.y;
  }
}

// ---------------------------------------------------------------------------
extern "C" void kernel_launch(void* const* d_in, const int* in_sizes, int n_in,
                              void* d_out, int out_size, void* d_ws, size_t ws_size,
                              hipStream_t stream) {
  const float* x     = (const float*)d_in[0];
  const float* w1    = (const float*)d_in[1];
  const float* gamma = (const float*)d_in[2];
  const float* beta  = (const float*)d_in[3];
  const float* mean  = (const float*)d_in[4];
  const float* var   = (const float*)d_in[5];
  const float* w2    = (const float*)d_in[6];
  const float* b2    = (const float*)d_in[7];
  const float* bias  = (const float*)d_in[8];
  float* out = (float*)d_out;

  float* pooled = (float*)d_ws;            // [B, DIM]        = 1536 f32
  float* ypad   = pooled + B_ * DIM_;      // [16, HID]       = 768 f32 (zero-padded)
  float* wdyn   = ypad + 16 * HID_;        // [B, DIM*49]     = 75264 f32

  pool_kernel<<<B_ * DIM_, 256, 0, stream>>>(x, pooled);
  se_fc1_kernel<<<(16 * HID_ + 255) / 256, 256, 0, stream>>>(
      pooled, w1, gamma, beta, mean, var, ypad);
  se_fc2_wmma_kernel<<<NOUT_ / 16, 32, 0, stream>>>(ypad, w2, b2, wdyn);
  dwconv_kernel<<<dim3(B_ * DIM_, H_ / TILE_H), 256, 0, stream>>>(
      x, wdyn, bias, out);
}